// CogAttn_29051158790664
// MI455X (gfx1250) — compile-verified
//
#include <hip/hip_runtime.h>
#include <hip/hip_bf16.h>
#include <math.h>

typedef __attribute__((ext_vector_type(2))) float v2f;
typedef __attribute__((ext_vector_type(8))) float v8f;

#define BB 32
#define CC 64
#define HH 8
#define TT 4000
#define PP 50
#define QQ 80        // TT/PP
#define TP 78        // QQ-2
#define FF 624       // HH*TP
#define NN (HH*TT)   // 32000 columns of the final einsum

// ---------------- Kernel 1: AvgPool2d((1,50)) ----------------
// One block per (b,c,h) row: coalesced load of 4000 floats to LDS, 80 window means.
__global__ void k_pool(const float* __restrict__ x, float* __restrict__ pooled) {
    __shared__ float sm[TT];
    const int row = blockIdx.x;                    // (b*CC + c)*HH + h
    const float* xr = x + (size_t)row * TT;
    for (int i = threadIdx.x; i < TT; i += blockDim.x) sm[i] = xr[i];
    __syncthreads();
    for (int q = threadIdx.x; q < QQ; q += blockDim.x) {
        float s = 0.f;
        const float* p = sm + q * PP;
        #pragma unroll
        for (int j = 0; j < PP; ++j) s += p[j];
        pooled[(size_t)row * QQ + q] = s * (1.0f / PP);
    }
}

// ---------------- Kernel 2: Conv2d(C,C,(1,3)) valid, + bias ----------------
// One block per (b,o,h); 78 threads along t'.
__global__ void k_conv(const float* __restrict__ pooled, const float* __restrict__ w,
                       const float* __restrict__ bias, float* __restrict__ repr) {
    const int h = blockIdx.x & 7;
    const int o = (blockIdx.x >> 3) & 63;
    const int b = blockIdx.x >> 9;
    const int t = threadIdx.x;                     // 0..77
    float s = bias[o];
    const float* pb = pooled + (((size_t)b * CC) * HH + h) * QQ + t;
    const float* wb = w + (size_t)o * CC * 3;
    for (int i = 0; i < CC; ++i) {
        const float* p = pb + (size_t)i * HH * QQ;
        s += wb[3*i+0]*p[0] + wb[3*i+1]*p[1] + wb[3*i+2]*p[2];
    }
    repr[((size_t)b * CC + o) * FF + h * TP + t] = s;
}

// ---------------- Kernel 3: transpose lin_w (64,624) -> lwT (624,64) ----------------
__global__ void k_transpose(const float* __restrict__ lw, float* __restrict__ lwT) {
    lwT[(size_t)blockIdx.x * CC + threadIdx.x] =
        lw[(size_t)threadIdx.x * FF + blockIdx.x];
}

// ---------------- Kernel 4: scores = relu(repr @ lwT + b), softmax over rows ----------------
// One workgroup (4 waves) per batch. Wave w owns output columns o = 16w..16w+15 and
// computes all 4 row tiles with v_wmma_f32_16x16x4_f32 over K = 624.
__global__ void k_scores(const float* __restrict__ repr, const float* __restrict__ lwT,
                         const float* __restrict__ lb, float* __restrict__ attn) {
    __shared__ float sm[CC * CC];                  // scores[c][o]
    const int b    = blockIdx.x;
    const int tid  = threadIdx.x;
    const int wave = tid >> 5;
    const int lane = tid & 31;
    const int lo   = lane & 15;                    // N (and M-row) index within tile
    const int kh   = (lane >> 4) << 1;             // 0 for lanes 0-15, 2 for lanes 16-31
    const int o    = wave * 16 + lo;
    const float* rb = repr + (size_t)b * CC * FF;

    v8f acc0 = {}, acc1 = {}, acc2 = {}, acc3 = {};
    for (int k0 = 0; k0 < FF; k0 += 4) {
        const int ka = k0 + kh;
        v2f bf;
        bf.x = lwT[(size_t)ka * CC + o];
        bf.y = lwT[(size_t)(ka + 1) * CC + o];
        v2f a0 = *(const v2f*)(rb + (size_t)( 0 + lo) * FF + ka);
        v2f a1 = *(const v2f*)(rb + (size_t)(16 + lo) * FF + ka);
        v2f a2 = *(const v2f*)(rb + (size_t)(32 + lo) * FF + ka);
        v2f a3 = *(const v2f*)(rb + (size_t)(48 + lo) * FF + ka);
        acc0 = __builtin_amdgcn_wmma_f32_16x16x4_f32(false, a0, false, bf, (short)0, acc0, false, false);
        acc1 = __builtin_amdgcn_wmma_f32_16x16x4_f32(false, a1, false, bf, (short)0, acc1, false, false);
        acc2 = __builtin_amdgcn_wmma_f32_16x16x4_f32(false, a2, false, bf, (short)0, acc2, false, false);
        acc3 = __builtin_amdgcn_wmma_f32_16x16x4_f32(false, a3, false, bf, (short)0, acc3, false, false);
    }

    const float biasO  = lb[o];
    const int   rowoff = (lane >> 4) * 8;          // C/D layout: VGPR r -> rows r, r+8
    #pragma unroll
    for (int r = 0; r < 8; ++r) {
        sm[( 0 + r + rowoff) * CC + o] = fmaxf(acc0[r] + biasO, 0.f);
        sm[(16 + r + rowoff) * CC + o] = fmaxf(acc1[r] + biasO, 0.f);
        sm[(32 + r + rowoff) * CC + o] = fmaxf(acc2[r] + biasO, 0.f);
        sm[(48 + r + rowoff) * CC + o] = fmaxf(acc3[r] + biasO, 0.f);
    }
    __syncthreads();

    // softmax over the row (c) dimension, one thread per output column o
    if (tid < CC) {
        float mx = -1e30f;
        for (int c = 0; c < CC; ++c) mx = fmaxf(mx, sm[c * CC + tid]);
        float s = 0.f;
        for (int c = 0; c < CC; ++c) s += __expf(sm[c * CC + tid] - mx);
        const float inv = 1.0f / s;
        float* ab = attn + (size_t)b * CC * CC;
        for (int c = 0; c < CC; ++c) ab[c * CC + tid] = __expf(sm[c * CC + tid] - mx) * inv;
    }
}

// ---------------- Kernel 5: y[b] = attn[b] (64x64) @ x[b] (64x32000) ----------------
// Block = (64-column slab, batch), 4 waves. attn staged in LDS (A-fragments via ds reads);
// each x fragment feeds 4 WMMAs (all row tiles), so x is read from HBM exactly once.
__global__ void k_apply(const float* __restrict__ x, const float* __restrict__ attn,
                        float* __restrict__ y) {
    __shared__ float sm[CC * CC];
    const int b   = blockIdx.y;
    const int tid = threadIdx.x;
    for (int i = tid; i < CC * CC; i += blockDim.x) sm[i] = attn[(size_t)b * CC * CC + i];
    __syncthreads();

    const int wave = tid >> 5;
    const int lane = tid & 31;
    const int lo   = lane & 15;
    const int kh   = (lane >> 4) << 1;
    const int n    = blockIdx.x * 64 + wave * 16 + lo;     // column in 0..31999
    const float* xb = x + (size_t)b * CC * NN;

    v8f acc0 = {}, acc1 = {}, acc2 = {}, acc3 = {};
    for (int k0 = 0; k0 < CC; k0 += 4) {
        const int kb = k0 + kh;
        if (k0 < CC - 4)                                   // uniform guard: prefetch next K slice
            __builtin_prefetch(xb + (size_t)(kb + 4) * NN + n, 0, 0);
        v2f bf;
        bf.x = xb[(size_t)kb * NN + n];
        bf.y = xb[(size_t)(kb + 1) * NN + n];
        v2f a0 = *(const v2f*)(sm + ( 0 + lo) * CC + kb);  // ds_load_b64, 8B aligned
        v2f a1 = *(const v2f*)(sm + (16 + lo) * CC + kb);
        v2f a2 = *(const v2f*)(sm + (32 + lo) * CC + kb);
        v2f a3 = *(const v2f*)(sm + (48 + lo) * CC + kb);
        acc0 = __builtin_amdgcn_wmma_f32_16x16x4_f32(false, a0, false, bf, (short)0, acc0, false, false);
        acc1 = __builtin_amdgcn_wmma_f32_16x16x4_f32(false, a1, false, bf, (short)0, acc1, false, false);
        acc2 = __builtin_amdgcn_wmma_f32_16x16x4_f32(false, a2, false, bf, (short)0, acc2, false, false);
        acc3 = __builtin_amdgcn_wmma_f32_16x16x4_f32(false, a3, false, bf, (short)0, acc3, false, false);
    }

    const int rowoff = (lane >> 4) * 8;
    float* yb = y + (size_t)b * CC * NN;
    #pragma unroll
    for (int r = 0; r < 8; ++r) {
        yb[(size_t)( 0 + r + rowoff) * NN + n] = acc0[r];
        yb[(size_t)(16 + r + rowoff) * NN + n] = acc1[r];
        yb[(size_t)(32 + r + rowoff) * NN + n] = acc2[r];
        yb[(size_t)(48 + r + rowoff) * NN + n] = acc3[r];
    }
}

extern "C" void kernel_launch(void* const* d_in, const int* in_sizes, int n_in,
                              void* d_out, int out_size, void* d_ws, size_t ws_size,
                              hipStream_t stream) {
    const float* x      = (const float*)d_in[0];
    const float* conv_w = (const float*)d_in[1];
    const float* conv_b = (const float*)d_in[2];
    const float* lin_w  = (const float*)d_in[3];
    const float* lin_b  = (const float*)d_in[4];
    float* y = (float*)d_out;

    float* ws     = (float*)d_ws;
    float* pooled = ws;                                  // 32*64*8*80   = 1,310,720 f
    float* repr   = pooled + (size_t)BB * CC * HH * QQ;  // 32*64*624   = 1,277,952 f
    float* lwT    = repr   + (size_t)BB * CC * FF;       // 624*64      =    39,936 f
    float* attn   = lwT    + (size_t)FF * CC;            // 32*64*64    =   131,072 f
    (void)ws_size; (void)in_sizes; (void)n_in; (void)out_size;

    k_pool     <<<BB * CC * HH, 256, 0, stream>>>(x, pooled);
    k_conv     <<<BB * CC * HH, TP,  0, stream>>>(pooled, conv_w, conv_b, repr);
    k_transpose<<<FF, CC, 0, stream>>>(lin_w, lwT);
    k_scores   <<<BB, 128, 0, stream>>>(repr, lwT, lin_b, attn);
    k_apply    <<<dim3(NN / 64, BB), 128, 0, stream>>>(x, attn, y);
}